// SceneGraph_MONet_42460046689010
// MI455X (gfx1250) — compile-verified
//
#include <hip/hip_runtime.h>

typedef __attribute__((ext_vector_type(2))) float v2f;
typedef __attribute__((ext_vector_type(8))) float v8f;

#define WMMA_F32(a, b, c) \
  __builtin_amdgcn_wmma_f32_16x16x4_f32(false, (a), false, (b), (short)0, (c), false, false)

// ---------------------------------------------------------------------------
// Kernel 1: 8x8 block-mean downsample of (planes,56,56) -> (…,7,7).
// chanMajor=1 writes fds[c][b*49+pq] (K-major layout for WMMA B operand).
// ---------------------------------------------------------------------------
__global__ __launch_bounds__(512) void ds_kernel(const float* __restrict__ src,
                                                 float* __restrict__ dst,
                                                 int C, int chanMajor) {
  int plane = blockIdx.x;
  int tid = threadIdx.x;
  int cell = tid >> 3;   // 0..63 (49 used)
  int rl = tid & 7;      // row-within-block
  float sum = 0.f;
  if (cell < 49) {
    int p = cell / 7, q = cell - (cell / 7) * 7;
    const float* row = src + (size_t)plane * 3136 + (size_t)(p * 8 + rl) * 56 + q * 8;
    float4 v0 = *(const float4*)(row);
    float4 v1 = *(const float4*)(row + 4);
    sum = v0.x + v0.y + v0.z + v0.w + v1.x + v1.y + v1.z + v1.w;
  }
  sum += __shfl_down(sum, 4, 8);
  sum += __shfl_down(sum, 2, 8);
  sum += __shfl_down(sum, 1, 8);
  if (cell < 49 && rl == 0) {
    float mean = sum * (1.0f / 64.0f);
    if (chanMajor) {
      int b = plane / C, c = plane - b * C;
      dst[(size_t)c * 392 + b * 49 + cell] = mean;
    } else {
      dst[(size_t)plane * 49 + cell] = mean;
    }
  }
}

// ---------------------------------------------------------------------------
// Kernel 2: generic WMMA f32 GEMM  Y = W(MxK,ldw) @ X(KxN,ldx) [+bias] [+addend]
// One wave -> one 16x16 output tile.  grid.x = ceil(N/16), grid.y = ceil(M/128).
// ---------------------------------------------------------------------------
__global__ __launch_bounds__(256) void gemm_wmma(
    const float* __restrict__ W, int ldw, const float* __restrict__ X, int ldx,
    float* __restrict__ Y, int ldy, const float* __restrict__ bias,
    const float* __restrict__ addend, int M, int K, int N) {
  int wave = threadIdx.x >> 5;
  int mtile = blockIdx.y * 8 + wave;
  if (mtile * 16 >= M) return;  // wave-uniform: EXEC stays all-ones for WMMA
  int ntile = blockIdx.x;
  int lane = threadIdx.x & 31;
  int l16 = lane & 15;
  int half = lane >> 4;

  int mrow = mtile * 16 + l16;
  int ncol = ntile * 16 + l16;
  if (ncol >= N) ncol = N - 1;  // tail: duplicate compute, guarded store

  // A layout (16x4 f32): VGPR0 = K{0|2}, VGPR1 = K{1|3}, split at lane 16.
  const float* wp = W + (size_t)mrow * ldw + 2 * half;
  // B layout (4x16 f32): mirrored; column n in lane, K rows split at lane 16.
  const float* xp = X + (size_t)(2 * half) * ldx + ncol;

  v8f acc = {0.f, 0.f, 0.f, 0.f, 0.f, 0.f, 0.f, 0.f};
  for (int k = 0; k < K; k += 4) {
    v2f a, b;
    a.x = wp[k];
    a.y = wp[k + 1];
    b.x = xp[(size_t)k * ldx];
    b.y = xp[(size_t)(k + 1) * ldx];
    __builtin_prefetch(wp + k + 64, 0, 1);  // global_prefetch_b8
    acc = WMMA_F32(a, b, acc);
  }

  int nc = ntile * 16 + l16;
  if (nc < N) {
#pragma unroll
    for (int j = 0; j < 8; ++j) {
      int m = mtile * 16 + j + 8 * half;  // D layout: VGPR j -> M = j + 8*(lane/16)
      float v = acc[j];
      if (bias) v += bias[m];
      if (addend) v += addend[(size_t)m * ldy + nc];
      Y[(size_t)m * ldy + nc] = v;
    }
  }
}

// ---------------------------------------------------------------------------
// Kernel 3: broadcast obj_feat into channel-major ofT[o][(b*11+n)*49+pq]
//           obj_feat = Gm[b,o,pq]*m[b,n,pq] + Gc[b,o,pq]
// ---------------------------------------------------------------------------
__global__ __launch_bounds__(256) void expand_obj(const float* __restrict__ Gm,
                                                  const float* __restrict__ Gc,
                                                  const float* __restrict__ mds,
                                                  float* __restrict__ ofT) {
  int idx = blockIdx.x * 256 + threadIdx.x;  // exact: 256*4312 threads
  int o = idx / 4312, col = idx - o * 4312;
  int bn = col / 49, pq = col - bn * 49;
  int b = bn / 11;
  int gi = o * 392 + b * 49 + pq;
  ofT[idx] = Gm[gi] * mds[col] + Gc[gi];
}

// ---------------------------------------------------------------------------
// Kernel 4: fused head.  rows = (b,n[,n']) pairs; 12544-K GEMM vs WfcT, +bias,
// L2-normalize.  A-tile (16 rows x 196 k) built on the fly in LDS (ReLU fused,
// rel_feat recomposed from S/O/Xr/Yr/Zr/m -> the big rel_in tensor never exists).
// Block = 8 waves; each wave owns two 16-col output tiles (16 tiles = 256 outs).
// ---------------------------------------------------------------------------
__global__ __launch_bounds__(256) void fused_head(
    const float* __restrict__ Wfc,  // (256, 12544) row-major
    const float* __restrict__ bias, const float* __restrict__ ofT,
    const float* __restrict__ S, const float* __restrict__ O,
    const float* __restrict__ Xr, const float* __restrict__ Yr,
    const float* __restrict__ Zr, const float* __restrict__ mds,
    float* __restrict__ out, int nrows, int mode) {
  __shared__ float vt[16][200];    // A tile: 16 rows x 196 K (padded)
  __shared__ float outS[16][260];  // staged 16x256 outputs (padded)
  __shared__ float red[16][16];
  __shared__ float invn[16];

  int mtile = blockIdx.x;
  int tid = threadIdx.x;
  int wave = tid >> 5, lane = tid & 31;
  int l16 = lane & 15, half = lane >> 4;
  int nt0 = wave * 2, nt1 = nt0 + 1;

  const float* B0 = Wfc + (size_t)(nt0 * 16 + l16) * 12544 + 2 * half;
  const float* B1 = Wfc + (size_t)(nt1 * 16 + l16) * 12544 + 2 * half;

  v8f acc0 = {0.f, 0.f, 0.f, 0.f, 0.f, 0.f, 0.f, 0.f};
  v8f acc1 = {0.f, 0.f, 0.f, 0.f, 0.f, 0.f, 0.f, 0.f};

  for (int s = 0; s < 64; ++s) {  // 64 stages x 196 K = 12544
    // cooperative LDS fill of the A tile (ReLU'd feature vector chunk)
    for (int e = tid; e < 16 * 196; e += 256) {
      int row = e / 196, kk = e - row * 196;
      int o = s * 4 + kk / 49, pq = kk % 49;
      int grow = mtile * 16 + row;
      if (grow >= nrows) grow = nrows - 1;  // tail rows: duplicate, store-guarded
      float v;
      if (mode == 0) {
        v = ofT[(size_t)o * 4312 + (size_t)grow * 49 + pq];
      } else {
        int b = grow / 121, rem = grow - b * 121;
        int n = rem / 11, n2 = rem - n * 11;
        int c1 = (b * 11 + n) * 49 + pq;
        int c2 = (b * 11 + n2) * 49 + pq;
        int gb = o * 392 + b * 49 + pq;
        v = S[(size_t)o * 4312 + c1] + O[(size_t)o * 4312 + c2] + Xr[gb] +
            Yr[gb] * mds[c1] + Zr[gb] * mds[c2];
      }
      vt[row][kk] = fmaxf(v, 0.f);
    }
    __syncthreads();

    int kbase = s * 196;
    for (int kk = 0; kk < 196; kk += 4) {
      int ka = kk + 2 * half;
      v2f a, b0v, b1v;
      a.x = vt[l16][ka];
      a.y = vt[l16][ka + 1];
      b0v.x = B0[kbase + kk];
      b0v.y = B0[kbase + kk + 1];
      b1v.x = B1[kbase + kk];
      b1v.y = B1[kbase + kk + 1];
      acc0 = WMMA_F32(a, b0v, acc0);
      acc1 = WMMA_F32(a, b1v, acc1);
    }
    __syncthreads();
  }

  // stage D tiles (+bias) into LDS for the row-wise L2 norm
#pragma unroll
  for (int j = 0; j < 8; ++j) {
    int r = j + 8 * half;
    outS[r][nt0 * 16 + l16] = acc0[j] + bias[nt0 * 16 + l16];
    outS[r][nt1 * 16 + l16] = acc1[j] + bias[nt1 * 16 + l16];
  }
  __syncthreads();

  int p = tid >> 4, i = tid & 15;
  float sq = 0.f;
  for (int j = i; j < 256; j += 16) {
    float v = outS[p][j];
    sq += v * v;
  }
  red[p][i] = sq;
  __syncthreads();
  if (tid < 16) {
    float tot = 0.f;
#pragma unroll
    for (int j = 0; j < 16; ++j) tot += red[tid][j];
    invn[tid] = rsqrtf(tot);
  }
  __syncthreads();

  int grow = mtile * 16 + p;
  if (grow < nrows) {
    float sc = invn[p];
    for (int j = i; j < 256; j += 16) out[(size_t)grow * 256 + j] = outS[p][j] * sc;
  }
}

// ---------------------------------------------------------------------------
extern "C" void kernel_launch(void* const* d_in, const int* in_sizes, int n_in,
                              void* d_out, int out_size, void* d_ws, size_t ws_size,
                              hipStream_t stream) {
  const float* feat = (const float*)d_in[0];
  const float* masks = (const float*)d_in[1];
  const float* Wc = (const float*)d_in[2];
  const float* bc = (const float*)d_in[3];
  const float* Wr = (const float*)d_in[4];
  const float* br = (const float*)d_in[5];
  const float* Wof = (const float*)d_in[6];
  const float* bof = (const float*)d_in[7];
  const float* Wrf = (const float*)d_in[8];
  const float* brf = (const float*)d_in[9];
  const float* Wofc = (const float*)d_in[10];
  const float* bofc = (const float*)d_in[11];
  const float* Wrfc = (const float*)d_in[12];
  const float* brfc = (const float*)d_in[13];
  float* out = (float*)d_out;
  float* ws = (float*)d_ws;

  // workspace layout (float offsets)
  float* fds = ws + 0;        // (256, 392)  K-major feat_ds
  float* mds = ws + 100352;   // (88, 49)    mask means
  float* cds = ws + 104704;   // (256, 392)  ctx_ds
  float* rds = ws + 205056;   // (384, 392)  rel_ds
  float* Gm = ws + 355584;    // (256, 392)  mask-scaled part of obj_feat
  float* Gc = ws + 455936;    // (256, 392)  constant part of obj_feat
  float* ofT = ws + 556288;   // (256, 4312) obj_feat channel-major
  float* S = ws + 1660160;    // (256, 4312) W1 @ obj_feat
  float* O = ws + 2764032;    // (256, 4312) W2 @ obj_feat
  float* Xr = ws + 3867904;   // (256, 392)  W3 @ xr + brf
  float* Yr = ws + 3968256;   // (256, 392)  W4 @ yr
  float* Zr = ws + 4068608;   // (256, 392)  W5 @ zr

  // 1) downsample (block_mean commutes with all linear layers)
  ds_kernel<<<2048, 512, 0, stream>>>(feat, fds, 256, 1);
  ds_kernel<<<88, 512, 0, stream>>>(masks, mds, 11, 0);

  dim3 blk(256);
  // 2) ctx_ds / rel_ds on the 392 downsampled columns
  gemm_wmma<<<dim3(25, 2), blk, 0, stream>>>(Wc, 256, fds, 392, cds, 392, bc, nullptr, 256, 256, 392);
  gemm_wmma<<<dim3(25, 3), blk, 0, stream>>>(Wr, 256, fds, 392, rds, 392, br, nullptr, 384, 256, 392);
  // 3) obj_feat = Gm*m + Gc :  Gm = Wof[:, :256]@fds + Wof[:,384:]@y ; Gc = Wof[:,256:384]@x + bof
  gemm_wmma<<<dim3(25, 2), blk, 0, stream>>>(Wof, 512, fds, 392, Gm, 392, nullptr, nullptr, 256, 256, 392);
  gemm_wmma<<<dim3(25, 2), blk, 0, stream>>>(Wof + 384, 512, cds + 128 * 392, 392, Gm, 392, nullptr, Gm, 256, 128, 392);
  gemm_wmma<<<dim3(25, 2), blk, 0, stream>>>(Wof + 256, 512, cds, 392, Gc, 392, bof, nullptr, 256, 128, 392);
  expand_obj<<<4312, 256, 0, stream>>>(Gm, Gc, mds, ofT);
  // 4) rel_feat decomposition pieces (brf folded into Xr)
  gemm_wmma<<<dim3(270, 2), blk, 0, stream>>>(Wrf, 896, ofT, 4312, S, 4312, nullptr, nullptr, 256, 256, 4312);
  gemm_wmma<<<dim3(270, 2), blk, 0, stream>>>(Wrf + 256, 896, ofT, 4312, O, 4312, nullptr, nullptr, 256, 256, 4312);
  gemm_wmma<<<dim3(25, 2), blk, 0, stream>>>(Wrf + 512, 896, rds, 392, Xr, 392, brf, nullptr, 256, 128, 392);
  gemm_wmma<<<dim3(25, 2), blk, 0, stream>>>(Wrf + 640, 896, rds + 128 * 392, 392, Yr, 392, nullptr, nullptr, 256, 128, 392);
  gemm_wmma<<<dim3(25, 2), blk, 0, stream>>>(Wrf + 768, 896, rds + 256 * 392, 392, Zr, 392, nullptr, nullptr, 256, 128, 392);
  // 5) fused ReLU + fc head + L2 norm  (obj: 88 rows, rel: 968 rows)
  fused_head<<<6, 256, 0, stream>>>(Wofc, bofc, ofT, S, O, Xr, Yr, Zr, mds, out, 88, 0);
  fused_head<<<61, 256, 0, stream>>>(Wrfc, brfc, ofT, S, O, Xr, Yr, Zr, mds, out + 88 * 256, 968, 1);
}